// ObstacleAwareAeroDMLoss_69492570849790
// MI455X (gfx1250) — compile-verified
//
#include <hip/hip_runtime.h>
#include <math.h>

typedef __attribute__((ext_vector_type(2))) float v2f;
typedef __attribute__((ext_vector_type(8))) float v8f;

#define BQ 64
#define SQ 4096
#define DQ 32
#define KQ 32
#define SCHUNK 128                      // s rows per block (8 waves x 16)
#define NCHUNK (SQ / SCHUNK)            // 32
#define NBLK (BQ * NCHUNK)              // 2048
#define NWAVE 8

// Weights
#define POS_W   2.0f
#define VEL_W   0.5f
#define LASTP_W 10.0f
#define OBS_W   10.0f

// Partial-sum slots in d_ws: [0]=pos, [1]=other, [2]=last, [3]=vel, [4]=pen2
__global__ __launch_bounds__(256)
void ObstacleLoss_partial(const float* __restrict__ pred,
                          const float* __restrict__ gt,
                          const float* __restrict__ centers,
                          const float* __restrict__ radii,
                          const float* __restrict__ mean,
                          const float* __restrict__ stdv,
                          float* __restrict__ ws)
{
    __shared__ float lds_pos[NWAVE][16][4];   // per-wave de-normalized xyz per row
    __shared__ float lds_part[NWAVE][5];

    const int tid  = (int)threadIdx.x;
    const int lane = tid & 31;
    const int w    = tid >> 5;
    const int blk  = (int)blockIdx.x;
    const int b    = blk / NCHUNK;
    const int chnk = blk % NCHUNK;
    const int s0   = chnk * SCHUNK + w * 16;

    // lane == channel
    const float mval = mean[lane];
    const float sval = stdv[lane];
    const bool  is_pos = (lane >= 1 && lane <= 3);

    float pos_sum = 0.f, other_sum = 0.f, last_sum = 0.f, vel_sum = 0.f, pen2_sum = 0.f;
    float prev_d = 0.f;

    // Read rows s0 .. s0+16 (the extra row only feeds velocity pairs).
    const int imax = (s0 + 16 < SQ) ? 16 : 15;
    for (int i = 0; i <= imax; ++i) {
        const int s = s0 + i;
        const size_t idx = ((size_t)b * SQ + s) * DQ + lane;
        const float p = pred[idx];
        const float g = gt[idx];
        const float d = p - g;
        const float d2 = d * d;
        if (i < 16) {
            if (is_pos) {
                pos_sum += d2;
                if (s == SQ - 1) last_sum += d2;
                lds_pos[w][i][lane - 1] = p * sval + mval;  // de-normalized position
            } else {
                other_sum += d2;
            }
        }
        if (i > 0 && is_pos) {
            const float vd = d - prev_d;     // (pred-gt)[s] - (pred-gt)[s-1]
            vel_sum += vd * vd;
        }
        prev_d = d;
    }

    __syncthreads();

    // -------- obstacle phase: d2 tiles via chained V_WMMA_F32_16X16X4_F32 --------
    // A (16x4): row M = (x,y,z,1); lanes 0-15 hold K=0,1; lanes 16-31 hold K=2,3.
    // B (4x16): col N = (-2cx,-2cy,-2cz,|c|^2); same K split, column = lane%16.
    // Second WMMA adds |p|^2 per row: A2 row = (|p|^2,0,0,0), B2 col = (1,0,0,0).
    const int l16  = lane & 15;
    const int half = lane >> 4;

    v2f A, A2, B2;
    {
        const float x = lds_pos[w][l16][0];
        const float y = lds_pos[w][l16][1];
        const float z = lds_pos[w][l16][2];
        if (half == 0) {
            A.x = x; A.y = y;
            A2.x = x * x + y * y + z * z; A2.y = 0.f;
            B2.x = 1.0f; B2.y = 0.f;
        } else {
            A.x = z; A.y = 1.0f;
            A2.x = 0.f; A2.y = 0.f;
            B2.x = 0.f; B2.y = 0.f;
        }
    }

    #pragma unroll
    for (int t = 0; t < 2; ++t) {
        const int obs = t * 16 + l16;
        const size_t cbase = ((size_t)b * KQ + obs) * 3;
        const float cx = centers[cbase + 0];
        const float cy = centers[cbase + 1];
        const float cz = centers[cbase + 2];
        v2f Bm;
        if (half == 0) { Bm.x = -2.f * cx; Bm.y = -2.f * cy; }
        else           { Bm.x = -2.f * cz; Bm.y = cx * cx + cy * cy + cz * cz; }

        v8f acc = {};
        // acc = A x Bm  (= -2 p.c + |c|^2)
        acc = __builtin_amdgcn_wmma_f32_16x16x4_f32(
            false, A,  false, Bm, (short)0, acc, false, false);
        // acc = A2 x B2 + acc  (= |p|^2 - 2 p.c + |c|^2 = d^2)
        acc = __builtin_amdgcn_wmma_f32_16x16x4_f32(
            false, A2, false, B2, (short)0, acc, false, false);

        const float thr = 2.2f * radii[(size_t)b * KQ + obs];
        #pragma unroll
        for (int j = 0; j < 8; ++j) {
            const float dd   = fmaxf(acc[j], 0.f);
            // raw v_sqrt_f32: inputs are O(1..100) squared distances, never
            // denormal, so skip the IEEE denorm-scaling fixup sequence.
            const float dist = __builtin_amdgcn_sqrtf(dd);
            const float pen  = fmaxf(thr - dist, 0.f);
            pen2_sum += pen * pen;
        }
    }

    // -------- wave32 reduction of the 5 partial sums --------
    #pragma unroll
    for (int off = 16; off > 0; off >>= 1) {
        pos_sum   += __shfl_down(pos_sum,   off, 32);
        other_sum += __shfl_down(other_sum, off, 32);
        last_sum  += __shfl_down(last_sum,  off, 32);
        vel_sum   += __shfl_down(vel_sum,   off, 32);
        pen2_sum  += __shfl_down(pen2_sum,  off, 32);
    }
    if (lane == 0) {
        lds_part[w][0] = pos_sum;
        lds_part[w][1] = other_sum;
        lds_part[w][2] = last_sum;
        lds_part[w][3] = vel_sum;
        lds_part[w][4] = pen2_sum;
    }
    __syncthreads();
    if (tid == 0) {
        float a0 = 0.f, a1 = 0.f, a2 = 0.f, a3 = 0.f, a4 = 0.f;
        #pragma unroll
        for (int q = 0; q < NWAVE; ++q) {
            a0 += lds_part[q][0]; a1 += lds_part[q][1]; a2 += lds_part[q][2];
            a3 += lds_part[q][3]; a4 += lds_part[q][4];
        }
        ws[0 * NBLK + blk] = a0;
        ws[1 * NBLK + blk] = a1;
        ws[2 * NBLK + blk] = a2;
        ws[3 * NBLK + blk] = a3;
        ws[4 * NBLK + blk] = a4;
    }
}

__global__ __launch_bounds__(256)
void ObstacleLoss_finalize(const float* __restrict__ ws, float* __restrict__ out)
{
    __shared__ float red[5][256];
    float s[5] = {0.f, 0.f, 0.f, 0.f, 0.f};
    for (int i = (int)threadIdx.x; i < NBLK; i += 256) {
        #pragma unroll
        for (int q = 0; q < 5; ++q) s[q] += ws[q * NBLK + i];
    }
    #pragma unroll
    for (int q = 0; q < 5; ++q) red[q][threadIdx.x] = s[q];
    __syncthreads();
    for (int stride = 128; stride > 0; stride >>= 1) {
        if ((int)threadIdx.x < stride) {
            #pragma unroll
            for (int q = 0; q < 5; ++q)
                red[q][threadIdx.x] += red[q][threadIdx.x + stride];
        }
        __syncthreads();
    }
    if (threadIdx.x == 0) {
        const float inv_bs = 1.0f / (float)(BQ * SQ);
        const float pos_mse    = red[0][0] * inv_bs;
        const float other_loss = red[1][0] * inv_bs / 29.0f;
        const float last_mse   = red[2][0] / (float)BQ;
        const float vel_loss   = red[3][0] / (float)(BQ * (SQ - 1));
        const float obst_loss  = red[4][0] * inv_bs;
        const float position_loss = pos_mse + LASTP_W * last_mse;
        const float total = POS_W * position_loss + VEL_W * vel_loss
                          + other_loss + OBS_W * obst_loss;
        out[0] = total;
        out[1] = position_loss;
        out[2] = vel_loss;
        out[3] = obst_loss;
    }
}

extern "C" void kernel_launch(void* const* d_in, const int* in_sizes, int n_in,
                              void* d_out, int out_size, void* d_ws, size_t ws_size,
                              hipStream_t stream) {
    const float* pred    = (const float*)d_in[0];
    const float* gt      = (const float*)d_in[1];
    const float* centers = (const float*)d_in[2];
    const float* radii   = (const float*)d_in[3];
    const float* mean    = (const float*)d_in[4];
    const float* stdv    = (const float*)d_in[5];
    float* ws  = (float*)d_ws;
    float* out = (float*)d_out;

    ObstacleLoss_partial<<<NBLK, 256, 0, stream>>>(pred, gt, centers, radii, mean, stdv, ws);
    ObstacleLoss_finalize<<<1, 256, 0, stream>>>(ws, out);
}